// CustomEmbeddingNet5_61014305406995
// MI455X (gfx1250) — compile-verified
//
#include <hip/hip_runtime.h>
#include <hip/hip_bf16.h>

// ---------------------------------------------------------------------------
// GCN (2x GCNConv + mean-pool + MLP) for MI455X (gfx1250, wave32).
//  - Edge scatter: one wave per edge, lane == channel (32 channels), native
//    fp32 global atomics (unsafeAtomicAdd -> global_atomic_add_f32).
//  - Dense transforms (h1 @ W2 and pooled @ Wf1) use V_WMMA_F32_16X16X4_F32
//    (exact fp32 math, matches the fp32 reference).
// ---------------------------------------------------------------------------

typedef __attribute__((ext_vector_type(2))) float v2f;
typedef __attribute__((ext_vector_type(8))) float v8f;

#define HIDDEN 32

__global__ void zero_f32(float* __restrict__ p, int n) {
  int i = blockIdx.x * blockDim.x + threadIdx.x;
  if (i < n) p[i] = 0.0f;
}

// deg[n] starts at 0; count in-edges.
__global__ void degree_count(const int* __restrict__ dst, int E, float* __restrict__ deg) {
  int e = blockIdx.x * blockDim.x + threadIdx.x;
  if (e < E) unsafeAtomicAdd(&deg[dst[e]], 1.0f);
}

// dinv[n] = rsqrt(deg[n] + 1)   (+1 = self loop; always > 0)
__global__ void make_dinv(float* __restrict__ deg_dinv, int N) {
  int n = blockIdx.x * blockDim.x + threadIdx.x;
  if (n < N) deg_dinv[n] = rsqrtf(deg_dinv[n] + 1.0f);
}

// Layer-1 edge message: agg[d][c] += dinv[s]*dinv[d] * x[s] * W1[c]
// (fuses x @ W1 since x has 1 feature). One wave per edge, lane = channel.
__global__ void edge_scatter_l1(const int* __restrict__ src, const int* __restrict__ dst,
                                const float* __restrict__ x, const float* __restrict__ dinv,
                                const float* __restrict__ W1, float* __restrict__ agg, int E) {
  int gid = blockIdx.x * blockDim.x + threadIdx.x;
  int e = gid >> 5;
  int c = gid & 31;
  if (e >= E) return;
  int s = src[e], d = dst[e];
  float norm = dinv[s] * dinv[d];
  unsafeAtomicAdd(&agg[(size_t)d * HIDDEN + c], norm * x[s] * W1[c]);
}

// Layer-2 edge message: agg[d][c] += dinv[s]*dinv[d] * xw[s][c]
__global__ void edge_scatter_l2(const int* __restrict__ src, const int* __restrict__ dst,
                                const float* __restrict__ xw, const float* __restrict__ dinv,
                                float* __restrict__ agg, int E) {
  int gid = blockIdx.x * blockDim.x + threadIdx.x;
  int e = gid >> 5;
  int c = gid & 31;
  if (e >= E) return;
  int s = src[e], d = dst[e];
  float norm = dinv[s] * dinv[d];
  unsafeAtomicAdd(&agg[(size_t)d * HIDDEN + c], norm * xw[(size_t)s * HIDDEN + c]);
}

// h1[n][c] = relu(agg[n][c] + dinv[n]^2 * x[n]*W1[c] + b1[c])
__global__ void post1(const float* __restrict__ agg, const float* __restrict__ x,
                      const float* __restrict__ dinv, const float* __restrict__ W1,
                      const float* __restrict__ b1, float* __restrict__ h1, int N) {
  int gid = blockIdx.x * blockDim.x + threadIdx.x;
  int n = gid >> 5;
  int c = gid & 31;
  if (n >= N) return;
  float di = dinv[n];
  float v = agg[(size_t)n * HIDDEN + c] + di * di * x[n] * W1[c] + b1[c];
  h1[(size_t)n * HIDDEN + c] = fmaxf(v, 0.0f);
}

// h2[n][c] = relu(agg[n][c] + dinv[n]^2 * xw[n][c] + b2[c]); written in-place over xw.
__global__ void post2(const float* __restrict__ agg, float* __restrict__ xw,
                      const float* __restrict__ dinv, const float* __restrict__ b2, int N) {
  int gid = blockIdx.x * blockDim.x + threadIdx.x;
  int n = gid >> 5;
  int c = gid & 31;
  if (n >= N) return;
  float di = dinv[n];
  size_t idx = (size_t)n * HIDDEN + c;
  float v = agg[idx] + di * di * xw[idx] + b2[c];
  xw[idx] = fmaxf(v, 0.0f);
}

// ---------------------------------------------------------------------------
// fp32 WMMA GEMM: C[rows x Nc] = A[rows x K] @ B[K x Nc] (+bias)(relu)
// rows % 16 == 0, K % 4 == 0, Nc % 16 == 0. One wave per 16x16 output tile.
// A frag (16x4):  lane l, vgpr v -> (M = l%16, K = 2*(l/16)+v)
// B frag (4x16):  lane l, vgpr v -> (K = 2*(l/16)+v, N = l%16)
// C/D (16x16):    lane l, vgpr v -> (M = v + 8*(l/16), N = l%16)
// ---------------------------------------------------------------------------
__global__ void wmma_gemm_f32(const float* __restrict__ A, const float* __restrict__ B,
                              const float* __restrict__ bias, float* __restrict__ C,
                              int rows, int K, int Nc, int relu) {
  const int lane = threadIdx.x & 31;
  const int wave = threadIdx.x >> 5;
  const int tilesPerRow = Nc >> 4;
  const int totalTiles = (rows >> 4) * tilesPerRow;
  int tile = blockIdx.x * (blockDim.x >> 5) + wave;
  if (tile >= totalTiles) return;             // wave-uniform: EXEC stays all-ones
  const int rowTile = tile / tilesPerRow;
  const int colTile = tile % tilesPerRow;
  const int r0 = rowTile << 4;
  const int c0 = colTile << 4;
  const int lm = lane & 15;                   // 0..15
  const int lh = lane >> 4;                   // 0 or 1

  v8f acc = {};
  for (int kk = 0; kk < K; kk += 4) {
    const float* ap = A + (size_t)(r0 + lm) * K + kk + 2 * lh;
    v2f a;
    a.x = ap[0];
    a.y = ap[1];
    const float* bp = B + (size_t)(kk + 2 * lh) * Nc + c0 + lm;
    v2f b;
    b.x = bp[0];
    b.y = bp[Nc];
    acc = __builtin_amdgcn_wmma_f32_16x16x4_f32(
        /*neg_a=*/false, a, /*neg_b=*/false, b,
        /*c_mod=*/(short)0, acc, /*reuse_a=*/false, /*reuse_b=*/false);
  }

#pragma unroll
  for (int v = 0; v < 8; ++v) {
    int m = v + 8 * lh;
    float val = acc[v];
    int col = c0 + lm;
    if (bias) val += bias[col];
    if (relu) val = fmaxf(val, 0.0f);
    C[(size_t)(r0 + m) * Nc + col] = val;
  }
}

// sums[g][c] += h2[n][c];  cnt[g] += 1 (lane 0 of each node-wave)
__global__ void pool_accum(const float* __restrict__ h2, const int* __restrict__ batch,
                           float* __restrict__ sums, float* __restrict__ cnt, int N) {
  int gid = blockIdx.x * blockDim.x + threadIdx.x;
  int n = gid >> 5;
  int c = gid & 31;
  if (n >= N) return;
  int g = batch[n];
  unsafeAtomicAdd(&sums[(size_t)g * HIDDEN + c], h2[(size_t)n * HIDDEN + c]);
  if (c == 0) unsafeAtomicAdd(&cnt[g], 1.0f);
}

__global__ void pool_finish(const float* __restrict__ sums, const float* __restrict__ cnt,
                            float* __restrict__ pooled, int G) {
  int gid = blockIdx.x * blockDim.x + threadIdx.x;
  int g = gid >> 5;
  int c = gid & 31;
  if (g >= G) return;
  pooled[(size_t)g * HIDDEN + c] = sums[(size_t)g * HIDDEN + c] / fmaxf(cnt[g], 1.0f);
}

// out[g][o] = sum_k act[g][k] * Wf2[k][o] + bf2[o]   (K=128, o<2)
__global__ void fc2_kernel(const float* __restrict__ act, const float* __restrict__ Wf2,
                           const float* __restrict__ bf2, float* __restrict__ out, int G) {
  int gid = blockIdx.x * blockDim.x + threadIdx.x;
  int g = gid >> 1;
  int o = gid & 1;
  if (g >= G) return;
  float acc = bf2[o];
  for (int k = 0; k < 128; ++k) acc += act[(size_t)g * 128 + k] * Wf2[k * 2 + o];
  out[(size_t)g * 2 + o] = acc;
}

extern "C" void kernel_launch(void* const* d_in, const int* in_sizes, int n_in,
                              void* d_out, int out_size, void* d_ws, size_t ws_size,
                              hipStream_t stream) {
  const float* x    = (const float*)d_in[0];
  const int*   ei   = (const int*)d_in[1];
  const int*   bat  = (const int*)d_in[2];
  const float* W1   = (const float*)d_in[3];
  const float* b1   = (const float*)d_in[4];
  const float* W2   = (const float*)d_in[5];
  const float* b2   = (const float*)d_in[6];
  const float* Wf1  = (const float*)d_in[7];
  const float* bf1  = (const float*)d_in[8];
  const float* Wf2  = (const float*)d_in[9];
  const float* bf2  = (const float*)d_in[10];
  float* out = (float*)d_out;

  const int N = in_sizes[0];            // 100000 (divisible by 16)
  const int E = in_sizes[1] / 2;        // 1600000
  const int G = out_size / 2;           // 1024
  const int* src = ei;
  const int* dst = ei + E;

  // Workspace layout (floats)
  float* ws = (float*)d_ws;
  float* buf0   = ws;                          // N*32 : agg1 -> xw2 -> h2
  float* buf1   = buf0 + (size_t)N * HIDDEN;   // N*32 : h1 -> agg2
  float* dinv   = buf1 + (size_t)N * HIDDEN;   // N    : deg -> dinv
  float* sums   = dinv + N;                    // G*32
  float* cnt    = sums + (size_t)G * HIDDEN;   // G
  float* pooled = cnt + G;                     // G*32
  float* fc1out = pooled + (size_t)G * HIDDEN; // G*128

  const int B = 256;
  auto blocks = [](long long n, int b) { return (unsigned)((n + b - 1) / b); };

  // 1) degrees -> dinv
  zero_f32<<<blocks(N, B), B, 0, stream>>>(dinv, N);
  degree_count<<<blocks(E, B), B, 0, stream>>>(dst, E, dinv);
  make_dinv<<<blocks(N, B), B, 0, stream>>>(dinv, N);

  // 2) layer 1: scatter fused x@W1 messages, then self-loop + bias + relu
  zero_f32<<<blocks((long long)N * HIDDEN, B), B, 0, stream>>>(buf0, N * HIDDEN);
  edge_scatter_l1<<<blocks((long long)E * HIDDEN, B), B, 0, stream>>>(src, dst, x, dinv, W1, buf0, E);
  post1<<<blocks((long long)N * HIDDEN, B), B, 0, stream>>>(buf0, x, dinv, W1, b1, buf1, N);

  // 3) layer 2 transform: xw2 = h1 @ W2  (WMMA fp32), buf1 -> buf0
  {
    int tiles = (N >> 4) * (HIDDEN >> 4);
    wmma_gemm_f32<<<blocks(tiles, 8), B, 0, stream>>>(buf1, W2, nullptr, buf0, N, HIDDEN, HIDDEN, 0);
  }

  // 4) layer 2: scatter, then self-loop + bias + relu (h2 overwrites buf0)
  zero_f32<<<blocks((long long)N * HIDDEN, B), B, 0, stream>>>(buf1, N * HIDDEN);
  edge_scatter_l2<<<blocks((long long)E * HIDDEN, B), B, 0, stream>>>(src, dst, buf0, dinv, buf1, E);
  post2<<<blocks((long long)N * HIDDEN, B), B, 0, stream>>>(buf1, buf0, dinv, b2, N);

  // 5) global mean pool
  zero_f32<<<blocks(G * HIDDEN, B), B, 0, stream>>>(sums, G * HIDDEN);
  zero_f32<<<blocks(G, B), B, 0, stream>>>(cnt, G);
  pool_accum<<<blocks((long long)N * HIDDEN, B), B, 0, stream>>>(buf0, bat, sums, cnt, N);
  pool_finish<<<blocks(G * HIDDEN, B), B, 0, stream>>>(sums, cnt, pooled, G);

  // 6) MLP: fc1 (WMMA fp32, bias+relu), then fc2 (tiny)
  {
    int tiles = (G >> 4) * (128 >> 4);
    wmma_gemm_f32<<<blocks(tiles, 8), B, 0, stream>>>(pooled, Wf1, bf1, fc1out, G, HIDDEN, 128, 1);
  }
  fc2_kernel<<<blocks((long long)G * 2, B), B, 0, stream>>>(fc1out, Wf2, bf2, out, G);
}